// TorchOpenpose_63230508532351
// MI455X (gfx1250) — compile-verified
//
#include <hip/hip_runtime.h>
#include <hip/hip_bf16.h>

// ---------------------------------------------------------------------------
// OpenPose peak + PAF scoring for MI455X (gfx1250).
//
// HBM-bound (AI ~2.8 FLOP/B << machine balance at 23.3 TB/s); f32 precision
// required (blur values compared bit-exactly vs the 3x3 max-pool). Vertical
// Gaussian blur runs on the CDNA5 tensor path as banded-matrix x patch
// matmuls (V_WMMA_F32_16X16X4_F32), with all fragment loads hoisted ahead of
// the WMMA chain so the 7 matrix ops issue back-to-back with staggered waits.
//
// Dataflow (no d2d copy needed):
//   vblur_wmma : heatmap -> d_out[0..IMG)   (vertical blur parked in out)
//   hblur      : d_out   -> ws              (full blur)
//   peaks      : ws      -> d_out[0..IMG)   (final peak_scores)
//   paf_scores :         -> d_out[IMG..)
// ---------------------------------------------------------------------------

namespace {

constexpr int J_ = 26, H_ = 1080, W_ = 1920;
constexpr int K_ = 26, N_ = 32;
constexpr float THRE1 = 0.1f, PAF_THRES = 0.4f;
constexpr int HW_ = H_ * W_;
constexpr int TILES_X = W_ / 16;          // 120 (exact)
constexpr int TILES_Y = (H_ + 15) / 16;   // 68  (last tile partial in H)

typedef float v2f __attribute__((ext_vector_type(2)));
typedef float v8f __attribute__((ext_vector_type(8)));

__device__ __forceinline__ int reflect_h(int y) {
  // jnp.pad mode='reflect': -1 -> 1, H -> H-2 (no edge duplication)
  return y < 0 ? -y : (y >= H_ ? 2 * (H_ - 1) - y : y);
}
__device__ __forceinline__ int reflect_w(int x) {
  return x < 0 ? -x : (x >= W_ ? 2 * (W_ - 1) - x : x);
}

// ---------------------------------------------------------------------------
// Kernel 1: vertical 11-tap Gaussian blur via WMMA f32 16x16x4.
// One wave computes one 16x16 output tile of one channel:
//   D(16x16) = G(16x28) * X(28x16),  G[m][r] = gauss[r - m] (banded),
//   X[r][n]  = in[j, reflect(y0-5+r), x0+n]
// K accumulated in 7 chunks of 4. Weight gathers use a zero-padded 44-entry
// LDS table so no per-lane guard (no EXEC juggling; WMMA needs EXEC==~0).
// ---------------------------------------------------------------------------
__global__ void vblur_wmma(const float* __restrict__ in, float* __restrict__ outp) {
  // sgpad[i] = gauss[i-16] for i in [16,26], else 0. Indices used: [1,43].
  __shared__ float sgpad[44];
  if (threadIdx.x == 0) {
    float t[11]; float s = 0.f;
#pragma unroll
    for (int i = 0; i < 11; ++i) { float x = (float)i - 5.0f; t[i] = expf(-(x * x) / 18.0f); s += t[i]; }
    for (int i = 0; i < 44; ++i) sgpad[i] = (i >= 16 && i <= 26) ? t[i - 16] / s : 0.f;
  }
  __syncthreads();

  const int wave = blockIdx.x * (blockDim.x >> 5) + (threadIdx.x >> 5);
  const int lane = threadIdx.x & 31;
  const int hf = lane >> 4;     // lane half (0/1)
  const int ln = lane & 15;     // index within half (M for A-frag, N for B-frag)

  const int tilesPerJ = TILES_X * TILES_Y;
  const int j  = wave / tilesPerJ;          // grid sized so wave < J_*tilesPerJ
  const int tr = wave - j * tilesPerJ;
  const int ty = tr / TILES_X;
  const int tx = tr - ty * TILES_X;
  const int y0 = ty * 16, x0 = tx * 16;

  // --- hoist A-fragment weights: ISA 7.12.2 f32 16x4 A layout is
  //     VGPR = K&1, lane-half = K>>1  =>  a.x: K = 4*kc+2*hf, a.y: K+1; M = ln.
  float aw[14];
  {
    const int bi = 16 + 2 * hf - ln;        // in [1,18]; +4*kc+1 stays < 44
#pragma unroll
    for (int kc = 0; kc < 7; ++kc) {
      aw[2 * kc]     = sgpad[bi + 4 * kc];
      aw[2 * kc + 1] = sgpad[bi + 4 * kc + 1];
    }
  }

  // --- hoist B-fragment loads (patch rows y0-5+K, col x0+ln).
  const float* base = in + (size_t)j * HW_;
  float bw[14];
  if (y0 >= 5 && y0 + 22 < H_) {
    // interior tile (ty 1..66): no reflection; one base pointer + immediate
    // offsets (max 25*1920*4 = 192000 B, fits the signed-24b IOFFSET).
    const float* p2 = base + (size_t)(y0 - 5 + 2 * hf) * W_ + (x0 + ln);
#pragma unroll
    for (int kc = 0; kc < 7; ++kc) {
      bw[2 * kc]     = p2[(size_t)(4 * kc) * W_];
      bw[2 * kc + 1] = p2[(size_t)(4 * kc + 1) * W_];
    }
  } else {
    // boundary tile (ty 0, 67): reflected rows (wave-uniform branch)
#pragma unroll
    for (int kc = 0; kc < 7; ++kc) {
      const int yB = y0 - 5 + 4 * kc + 2 * hf;
      bw[2 * kc]     = base[(size_t)reflect_h(yB)     * W_ + (x0 + ln)];
      bw[2 * kc + 1] = base[(size_t)reflect_h(yB + 1) * W_ + (x0 + ln)];
    }
  }

  // --- 7 back-to-back WMMAs (accumulator chain)
  v8f c = {0.f, 0.f, 0.f, 0.f, 0.f, 0.f, 0.f, 0.f};
#pragma unroll
  for (int kc = 0; kc < 7; ++kc) {
    v2f a; a.x = aw[2 * kc]; a.y = aw[2 * kc + 1];
    v2f b; b.x = bw[2 * kc]; b.y = bw[2 * kc + 1];
    // (neg_a, A, neg_b, B, c_mod, C, reuse_a, reuse_b)
    c = __builtin_amdgcn_wmma_f32_16x16x4_f32(false, a, false, b, (short)0, c, false, false);
  }

  // --- store; D layout: VGPR r -> M = r + 8*hf, N = ln.
  float* ob = outp + (size_t)j * HW_;
  if (y0 + 16 <= H_) {
    // full tile (ty 0..66): unconditional store clause, immediate offsets
    // (max 7*1920*4 = 53760 B, fits IOFFSET) -- no exec masking.
    float* p = ob + (size_t)(y0 + 8 * hf) * W_ + (x0 + ln);
#pragma unroll
    for (int r = 0; r < 8; ++r) p[(size_t)r * W_] = c[r];
  } else {
    // partial tile (ty 67): guard rows >= H
#pragma unroll
    for (int r = 0; r < 8; ++r) {
      const int y = y0 + r + 8 * hf;
      if (y < H_) ob[(size_t)y * W_ + (x0 + ln)] = c[r];
    }
  }
}

// ---------------------------------------------------------------------------
// Kernel 2: horizontal 11-tap blur (streaming; L0/L2 absorb the 11x reuse)
// ---------------------------------------------------------------------------
__global__ void hblur(const float* __restrict__ in, float* __restrict__ outp) {
  const int idx = blockIdx.x * blockDim.x + threadIdx.x;
  if (idx >= J_ * HW_) return;
  float g[11]; float s = 0.f;
#pragma unroll
  for (int i = 0; i < 11; ++i) { float x = (float)i - 5.0f; g[i] = expf(-(x * x) / 18.0f); s += g[i]; }
  const float inv = 1.0f / s;
#pragma unroll
  for (int i = 0; i < 11; ++i) g[i] *= inv;   // constant indices -> VGPRs

  const int x = idx % W_;
  const float* row = in + (idx - x);
  float acc = 0.f;
#pragma unroll
  for (int t = 0; t < 11; ++t) acc += g[t] * row[reflect_w(x + t - 5)];
  outp[idx] = acc;
}

// ---------------------------------------------------------------------------
// Kernel 3: 3x3 SAME max-pool (-inf pad == clipped window) + local-max mask
// ---------------------------------------------------------------------------
__global__ void peaks(const float* __restrict__ blur, float* __restrict__ outp) {
  const int idx = blockIdx.x * blockDim.x + threadIdx.x;
  if (idx >= J_ * HW_) return;
  const int x = idx % W_;
  const int y = (idx / W_) % H_;
  const float* img = blur + ((size_t)idx - (size_t)y * W_ - x);
  const float v = blur[idx];
  float mx = v;
#pragma unroll
  for (int dy = -1; dy <= 1; ++dy)
#pragma unroll
    for (int dx = -1; dx <= 1; ++dx) {
      const int yy = y + dy, xx = x + dx;
      if (yy >= 0 && yy < H_ && xx >= 0 && xx < W_)
        mx = fmaxf(mx, img[(size_t)yy * W_ + xx]);
    }
  outp[idx] = (v == mx && v > THRE1) ? v : 0.f;
}

// ---------------------------------------------------------------------------
// Kernel 4: PAF line-integral scores (26x32x32 pairs, 10 samples each; tiny)
// ---------------------------------------------------------------------------
__global__ void paf_scores(const float* __restrict__ pafv,
                           const int* __restrict__ candA,
                           const int* __restrict__ candB,
                           float* __restrict__ outp) {
  const int idx = blockIdx.x * blockDim.x + threadIdx.x;
  if (idx >= K_ * N_ * N_) return;
  const int k = idx / (N_ * N_);
  const int rem = idx - k * (N_ * N_);
  const int i = rem >> 5;       // index into candA
  const int jj = rem & 31;      // index into candB
  const float ax = (float)candA[(k * N_ + i) * 2 + 0];
  const float ay = (float)candA[(k * N_ + i) * 2 + 1];
  const float bx = (float)candB[(k * N_ + jj) * 2 + 0];
  const float by = (float)candB[(k * N_ + jj) * 2 + 1];
  const float vx = bx - ax, vy = by - ay;
  const float nrm = sqrtf(vx * vx + vy * vy) + 1e-10f;
  const float ux = vx / nrm, uy = vy / nrm;
  const float* p0 = pafv + (size_t)(2 * k) * HW_;
  const float* p1 = pafv + (size_t)(2 * k + 1) * HW_;
  float s = 0.f;
#pragma unroll
  for (int m = 0; m < 10; ++m) {
    const float t = (float)m / 9.0f;        // jnp.linspace(0,1,10)
    const int xi = (int)(ax + vx * t);      // trunc toward zero == astype(int32)
    const int yi = (int)(ay + vy * t);
    const size_t off = (size_t)yi * W_ + xi;
    s += p0[off] * ux + p1[off] * uy;
  }
  s *= 0.1f;                                // mean over 10
  s = fminf(fmaxf(s, 0.f), 1.f);            // clip [0,1]
  outp[idx] = (s > PAF_THRES) ? s : 0.f;
}

} // namespace

extern "C" void kernel_launch(void* const* d_in, const int* in_sizes, int n_in,
                              void* d_out, int out_size, void* d_ws, size_t ws_size,
                              hipStream_t stream) {
  const float* heatmap = (const float*)d_in[0];
  const float* pafv    = (const float*)d_in[1];
  const int*   candA   = (const int*)d_in[2];
  const int*   candB   = (const int*)d_in[3];
  float* out = (float*)d_out;
  float* ws0 = (float*)d_ws;                      // J*H*W floats (215.7 MB)
  const size_t IMG = (size_t)J_ * HW_;

  // 1) vertical blur (WMMA): heatmap -> out (parked in the peak slot)
  //    26*68*120 = 212160 tiles; 8 waves per 256-thread block -> 26520 blocks (exact)
  vblur_wmma<<<(J_ * TILES_X * TILES_Y) / 8, 256, 0, stream>>>(heatmap, out);
  // 2) horizontal blur: out -> ws0 (ws0 now holds the full blur)
  hblur<<<(int)((IMG + 255) / 256), 256, 0, stream>>>(out, ws0);
  // 3) peaks: read blur from ws0, write final peak_scores -> out[0..IMG)
  peaks<<<(int)((IMG + 255) / 256), 256, 0, stream>>>(ws0, out);
  // 4) PAF pair scores -> out[IMG..)
  paf_scores<<<(K_ * N_ * N_ + 255) / 256, 256, 0, stream>>>(pafv, candA, candB, out + IMG);
}